// DEQLayer_36696200577499
// MI455X (gfx1250) — compile-verified
//
#include <hip/hip_runtime.h>
#include <hip/hip_bf16.h>
#include <math.h>

// Problem constants (match reference)
#define BB 524288
#define HH 128
#define NELEM (BB * HH)          // 67,108,864 floats per [B,H] matrix
#define MAX_ITER 100
#define TOLER 1e-5f

typedef float v2f __attribute__((ext_vector_type(2)));
typedef float v8f __attribute__((ext_vector_type(8)));

constexpr int ROWS_PER_BLK = 128;          // 8 waves x 16 rows
constexpr int NBLK = BB / ROWS_PER_BLK;    // 4096 blocks
constexpr int THREADS = 256;               // 8 wave32s

// ---------------------------------------------------------------------------
// Load a 128x128 f32 matrix (row-major, rows = output feature h, cols = k)
// into LDS with an XOR swizzle at float2 granularity:
//   pair p = k/2 stored at p ^ (h & 15)
// This makes the per-lane B-fragment ds_load_b64 (lanes 0..15 have distinct
// h mod 16) hit 16 distinct bank-pairs -> conflict free.
// ---------------------------------------------------------------------------
__device__ __forceinline__ void load_matrix_lds(float* sW, const float* Wg) {
  for (int i = threadIdx.x; i < 128 * 64; i += THREADS) {
    const int n = i >> 6;   // row (h)
    const int p = i & 63;   // pair index (k/2)
    const float2 w2 = *reinterpret_cast<const float2*>(Wg + n * 128 + 2 * p);
    const int ps = p ^ (n & 15);
    *reinterpret_cast<float2*>(sW + n * 128 + 2 * ps) = w2;
  }
}

// ---------------------------------------------------------------------------
// One 16-row-per-wave stripe of  out = [tanh]( A @ W^T + (ux | bias) )
// using V_WMMA_F32_16X16X4_F32.  A-frag layout (ISA 7.12.2, 32-bit A 16x4):
//   lane<16: m=lane, {k0,k0+1}; lane>=16: m=lane-16, {k0+2,k0+3}
// B-frag (4x16, KxN): lane<16: n=lane, {k0,k0+1}; lane>=16: n=lane-16,{k0+2,k0+3}
// Returns per-lane sum of squared diffs (if DIFF).
// ---------------------------------------------------------------------------
template <bool TANH, bool HAS_UX, bool DIFF>
__device__ __forceinline__ float stripe_gemm(const float* __restrict__ A_g,
                                             const float* __restrict__ uxOrBias,
                                             const float* sW,
                                             float* __restrict__ outp) {
  const int tid  = threadIdx.x;
  const int wave = tid >> 5;
  const int lane = tid & 31;
  const int half = lane >> 4;      // 0: K pair {0,1}, 1: K pair {2,3}
  const int l16  = lane & 15;
  const int row0 = blockIdx.x * ROWS_PER_BLK + wave * 16;

  // Load all 32 A fragments for this wave's 16 rows (K = 0..127).
  v2f a[32];
  const float* arow = A_g + (size_t)(row0 + l16) * HH + half * 2;
#pragma unroll
  for (int kt = 0; kt < 32; ++kt)
    a[kt] = *reinterpret_cast<const v2f*>(arow + kt * 4);

  float sumsq = 0.0f;

#pragma unroll 1
  for (int nt = 0; nt < 8; ++nt) {
    const int n = nt * 16 + l16;            // output column / W row
    // C/D layout (ISA 7.12.2): vgpr i, lane<16 -> M=i, lane>=16 -> M=i+8, N=l16
    v8f c0, c1;
    if (HAS_UX) {
#pragma unroll
      for (int i = 0; i < 8; ++i)
        c0[i] = uxOrBias[(size_t)(row0 + i + half * 8) * HH + n];
    } else {
      const float bias = uxOrBias[n];
#pragma unroll
      for (int i = 0; i < 8; ++i) c0[i] = bias;
    }
#pragma unroll
    for (int i = 0; i < 8; ++i) c1[i] = 0.0f;

    const float* wrow = sW + n * 128;
    // Two independent accumulation chains to halve the WMMA RAW chain length.
#pragma unroll
    for (int kt = 0; kt < 32; kt += 2) {
      const int p0 = kt * 2 + half;         // pair index of k0 for kt
      const int p1 = (kt + 1) * 2 + half;   // pair index for kt+1
      v2f b0 = *reinterpret_cast<const v2f*>(wrow + 2 * (p0 ^ l16));
      v2f b1 = *reinterpret_cast<const v2f*>(wrow + 2 * (p1 ^ l16));
      c0 = __builtin_amdgcn_wmma_f32_16x16x4_f32(false, a[kt], false, b0,
                                                 (short)0, c0, false, false);
      c1 = __builtin_amdgcn_wmma_f32_16x16x4_f32(false, a[kt + 1], false, b1,
                                                 (short)0, c1, false, false);
    }

#pragma unroll
    for (int i = 0; i < 8; ++i) {
      const size_t idx = (size_t)(row0 + i + half * 8) * HH + n;
      float v = c0[i] + c1[i];
      if (TANH) v = tanhf(v);
      if (DIFF) {
        const float d = v - A_g[idx];
        sumsq += d * d;
      }
      outp[idx] = v;
    }
  }
  return sumsq;
}

// ---------------------------------------------------------------------------
// Kernels
// ---------------------------------------------------------------------------
__global__ __launch_bounds__(THREADS) void deq_init(float* __restrict__ z0,
                                                    int* __restrict__ ctrl) {
  const size_t base = (size_t)blockIdx.x * ROWS_PER_BLK * HH;
  float4* dst = reinterpret_cast<float4*>(z0 + base);
  const float4 zero = make_float4(0.f, 0.f, 0.f, 0.f);
  for (int i = threadIdx.x; i < ROWS_PER_BLK * HH / 4; i += THREADS) dst[i] = zero;
  if (blockIdx.x == 0 && threadIdx.x == 0) { ctrl[0] = 0; ctrl[1] = 0; }
}

// ux = x @ U_w^T + U_b
__global__ __launch_bounds__(THREADS) void deq_inject(
    const float* __restrict__ x, const float* __restrict__ Uw,
    const float* __restrict__ Ub, float* __restrict__ ux) {
  __shared__ float sW[128 * 128];
  load_matrix_lds(sW, Uw);
  __syncthreads();
  stripe_gemm<false, false, false>(x, Ub, sW, ux);
}

// z_dst = tanh(z_src @ W^T + ux); per-wave squared-diff partials
__global__ __launch_bounds__(THREADS) void deq_step(
    const float* __restrict__ zsrc, float* __restrict__ zdst,
    const float* __restrict__ ux, const float* __restrict__ W,
    float* __restrict__ parts, const int* __restrict__ ctrl) {
  __shared__ float sW[128 * 128];
  if (ctrl[0]) return;  // already converged: uniform early-exit
  load_matrix_lds(sW, W);
  __syncthreads();
  float s = stripe_gemm<true, true, true>(zsrc, ux, sW, zdst);
#pragma unroll
  for (int o = 16; o > 0; o >>= 1) s += __shfl_xor(s, o, 32);
  if ((threadIdx.x & 31) == 0)
    parts[(size_t)blockIdx.x * 8 + (threadIdx.x >> 5)] = s;
}

// Deterministic fixed-order reduction of 4096*8 partials; updates ctrl.
__global__ void deq_reduce(const float* __restrict__ parts,
                           int* __restrict__ ctrl, int it) {
  if (ctrl[0]) return;
  __shared__ float r[256];
  float s = 0.0f;
  const float* p = parts + (size_t)threadIdx.x * 128;
  for (int i = 0; i < 128; ++i) s += p[i];   // fixed order
  r[threadIdx.x] = s;
  __syncthreads();
  for (int stride = 128; stride > 0; stride >>= 1) {
    if (threadIdx.x < stride) r[threadIdx.x] += r[threadIdx.x + stride];
    __syncthreads();
  }
  if (threadIdx.x == 0) {
    ctrl[1] = it + 1;                        // iterations completed
    if (r[0] < TOLER * TOLER) ctrl[0] = 1;   // ||dz||_F < TOL
  }
}

// z_out = converged ? tanh(z @ W^T + ux) : z
__global__ __launch_bounds__(THREADS) void deq_final(
    const float* __restrict__ z0, const float* __restrict__ z1,
    const float* __restrict__ ux, const float* __restrict__ W,
    float* __restrict__ out, const int* __restrict__ ctrl) {
  __shared__ float sW[128 * 128];
  const int done = ctrl[0];
  const int iters = ctrl[1];
  const float* zcur = (iters & 1) ? z1 : z0;  // ping-pong parity
  if (done) {
    load_matrix_lds(sW, W);
    __syncthreads();
    stripe_gemm<true, true, false>(zcur, ux, sW, out);
  } else {
    const size_t base = (size_t)blockIdx.x * ROWS_PER_BLK * HH;
    const float4* src = reinterpret_cast<const float4*>(zcur + base);
    float4* dst = reinterpret_cast<float4*>(out + base);
    for (int i = threadIdx.x; i < ROWS_PER_BLK * HH / 4; i += THREADS)
      dst[i] = src[i];
  }
}

__global__ void deq_meta(float* __restrict__ out, const int* __restrict__ ctrl) {
  if (threadIdx.x == 0 && blockIdx.x == 0) {
    out[(size_t)NELEM] = (float)ctrl[1];
    out[(size_t)NELEM + 1] = ctrl[0] ? 1.0f : 0.0f;
  }
}

// ---------------------------------------------------------------------------
// Host launch: fixed 204-kernel sequence (graph-capture safe, deterministic).
// Workspace layout (floats): ux[N] | z0[N] | z1[N] | parts[32768] | ctrl(2 int)
// Requires ws_size >= 3*N*4 + 32768*4 + 8 bytes (~805 MB).
// ---------------------------------------------------------------------------
extern "C" void kernel_launch(void* const* d_in, const int* in_sizes, int n_in,
                              void* d_out, int out_size, void* d_ws,
                              size_t ws_size, hipStream_t stream) {
  const float* x  = (const float*)d_in[0];  // [B, D]
  const float* W  = (const float*)d_in[1];  // [H, H]
  const float* Uw = (const float*)d_in[2];  // [H, D]
  const float* Ub = (const float*)d_in[3];  // [H]
  float* out = (float*)d_out;

  float* ws    = (float*)d_ws;
  float* ux    = ws;
  float* z0    = ws + (size_t)NELEM;
  float* z1    = ws + (size_t)2 * NELEM;
  float* parts = ws + (size_t)3 * NELEM;
  int*   ctrl  = (int*)(parts + NBLK * 8);

  const dim3 grid(NBLK), blk(THREADS);
  deq_init<<<grid, blk, 0, stream>>>(z0, ctrl);
  deq_inject<<<grid, blk, 0, stream>>>(x, Uw, Ub, ux);
  for (int it = 0; it < MAX_ITER; ++it) {
    const float* zs = (it & 1) ? z1 : z0;
    float*       zd = (it & 1) ? z0 : z1;
    deq_step<<<grid, blk, 0, stream>>>(zs, zd, ux, W, parts, ctrl);
    deq_reduce<<<1, 256, 0, stream>>>(parts, ctrl, it);
  }
  deq_final<<<grid, blk, 0, stream>>>(z0, z1, ux, W, out, ctrl);
  deq_meta<<<1, 1, 0, stream>>>(out, ctrl);
}